// LinearFilterKernel_36515811950715
// MI455X (gfx1250) — compile-verified
//
#include <hip/hip_runtime.h>
#include <hip/hip_bf16.h>

// ---------------------------------------------------------------------------
// Position-varying 5x5 filter (kernel-prediction style) on gfx1250 (MI455X).
//
// Per output pixel: D[12x9] = A[12x225] x B[225x9]  (+bias, clamp to [0,1])
//   M = (b,c) pairs  (4*3 = 12, padded to 16)
//   N = st_out       (9, padded to 16)
//   K = st_in * taps, reordered as i*28 + k (taps 25 padded to 28 = 7*4)
// One wave32 per pixel -> 63 chained V_WMMA_F32_16X16X4_F32 steps, fully
// unrolled. Problem is HBM-bound (~115 MB unique, ~0.45 GFLOP -> ~5 us floor
// at 23.3 TB/s), so native-f32 WMMA is the right matrix op; no downconvert.
//
// Divergence-free inner loop: D rows 12..15 / cols 9..15 are never stored,
// so their A/B contents are don't-care -> clamp addresses, no exec masking.
// Only real zeroing: tap-pad (k>=25) in B, top/left boundary taps in A.
// ---------------------------------------------------------------------------

typedef __attribute__((ext_vector_type(2))) float v2f;
typedef __attribute__((ext_vector_type(8))) float v8f;

#define ST   9
#define NC   3
#define IH   288
#define IW   288
#define KS   5
#define STR  3
#define OWW  96
#define K2   25
#define TSTEPS 7                    // 28 / 4 K-chunks per st_in plane
#define PIX  9216                   // 96*96
#define LF_I_STRIDE (NC * IH * IW)  // 248832 floats between st_in planes
#define W_OI_STRIDE (PIX * K2)      // 230400 floats between (o,i) planes

__global__ __launch_bounds__(256)
void kpn_wmma_f32_kernel(const float* __restrict__ lf,
                         const float* __restrict__ weights,
                         const float* __restrict__ biases,
                         float* __restrict__ out)
{
    const int lane = threadIdx.x & 31;
    const int wave = threadIdx.x >> 5;
    const int pix  = blockIdx.x * 8 + wave;     // 9216 pixels, exact fit
    const int oh   = pix / OWW;
    const int ow   = pix - OWW * oh;

    const int half = lane >> 4;                 // K-in-chunk = v + 2*half
    const int mn   = lane & 15;                 // A-row / B-col index

    // A row -> (b,c). Rows 12..15 clamped to 11 (their D rows never stored).
    const int mA = (mn < 12) ? mn : 11;
    const int bb = mA / 3;
    const int cc = mA - 3 * bb;
    const float* lfP = lf + (long)(bb * ST * NC + cc) * (IH * IW);

    // B col -> o. Cols 9..15 clamped to 8 (their D cols never stored).
    const int oB = (mn < 9) ? mn : 8;
    const float* wP = weights + (long)oB * ST * W_OI_STRIDE + (long)pix * K2;

    // pad: pl = pt = 2, pr = pb = 0  ->  y = oh*3 + kh - 2 in [-2, 287]
    const int y0 = oh * STR - 2;
    const int x0 = ow * STR - 2;

    // Warm the per-pixel weight stream (gfx1250 global_prefetch path).
    __builtin_prefetch(&wP[0], 0, 1);
    __builtin_prefetch(&wP[4 * (long)W_OI_STRIDE], 0, 1);

    // ---- precompute per-(t,v) tap offsets & masks once (i-independent) ---
    int  aOff[TSTEPS][2];   // clamped lf offset  y*IW + x
    int  kOff[TSTEPS][2];   // clamped tap index  (address-safe)
    bool aOk [TSTEPS][2];   // A value valid (tap real && y>=0 && x>=0)
    bool kOk [TSTEPS][2];   // B value valid (tap real)
#pragma unroll
    for (int t = 0; t < TSTEPS; ++t) {
#pragma unroll
        for (int v = 0; v < 2; ++v) {
            const int  k  = 4 * t + 2 * half + v;   // 0..27
            const bool kv = (k < K2);
            const int  kc = kv ? k : (K2 - 1);
            const int  kh = kc / KS;
            const int  kw = kc - KS * kh;
            const int  y  = y0 + kh;
            const int  x  = x0 + kw;
            kOk [t][v] = kv;
            kOff[t][v] = kc;
            aOk [t][v] = kv & (y >= 0) & (x >= 0);
            aOff[t][v] = (y >= 0 ? y : 0) * IW + (x >= 0 ? x : 0);
        }
    }

    // ---- main chain: 9 planes x 7 chunks = 63 WMMAs, no branches ---------
    v8f acc = {};
#pragma unroll
    for (int i = 0; i < ST; ++i) {
        const float* lfI = lfP + (long)i * LF_I_STRIDE;
        const float* wI  = wP  + (long)i * W_OI_STRIDE;
#pragma unroll
        for (int t = 0; t < TSTEPS; ++t) {
            v2f afrag, bfrag;
#pragma unroll
            for (int v = 0; v < 2; ++v) {
                const float av = lfI[aOff[t][v]];   // unconditional load
                const float bv = wI [kOff[t][v]];   // unconditional load
                afrag[v] = aOk[t][v] ? av : 0.0f;   // one v_cndmask each
                bfrag[v] = kOk[t][v] ? bv : 0.0f;
            }
            acc = __builtin_amdgcn_wmma_f32_16x16x4_f32(
                false, afrag, false, bfrag, (short)0, acc, false, false);
        }
    }

    // ---- epilogue: bias + clamp + scatter the valid 12x9 block -----------
    const float bias = biases[oB * PIX + pix];
#pragma unroll
    for (int r = 0; r < 8; ++r) {
        const int m = r + 8 * half;                 // (b,c) row of D
        if ((m < 12) & (mn < 9)) {
            const int ob = m / 3;
            const int oc = m - 3 * ob;
            float vv = acc[r] + bias;
            vv = fminf(fmaxf(vv, 0.0f), 1.0f);
            out[(long)((ob * ST + mn) * NC + oc) * PIX + pix] = vv;
        }
    }
}

extern "C" void kernel_launch(void* const* d_in, const int* in_sizes, int n_in,
                              void* d_out, int out_size, void* d_ws, size_t ws_size,
                              hipStream_t stream) {
    (void)in_sizes; (void)n_in; (void)d_ws; (void)ws_size; (void)out_size;
    const float* lf      = (const float*)d_in[0];
    const float* weights = (const float*)d_in[1];
    const float* biases  = (const float*)d_in[2];
    float* out = (float*)d_out;

    // 9216 pixels, 8 waves (one pixel each) per 256-thread block.
    dim3 grid(PIX / 8), block(256);
    kpn_wmma_f32_kernel<<<grid, block, 0, stream>>>(lf, weights, biases, out);
}